// HolographicMemory_59639915872449
// MI455X (gfx1250) — compile-verified
//
#include <hip/hip_runtime.h>
#include <math.h>

// ---------------------------------------------------------------------------
// HolographicMemory on MI455X (gfx1250, wave32).
//   k1: G = H^T H per head, split-K x4 (f32 WMMA 16x16x4, TDM double-buffered
//       LDS staging via tensor_load_to_lds, fallback: manual + prefetch)
//   k2: 64x64 symmetric Jacobi eig (sums split-K partials) -> bulk = S*Vh
//   k3: kk = bulk Wk^T, vv = bulk Wv^T        (f32 WMMA)
//   k4: q/scores/softmax/ctx/out/LayerNorm fused per 16-row tile (f32 WMMA)
// ---------------------------------------------------------------------------

typedef float v2f __attribute__((ext_vector_type(2)));
typedef float v8f __attribute__((ext_vector_type(8)));

#define BH      128
#define NSEQ    2048
#define DH      64
#define MROWS   512
#define KMODES  32
#define SPLITK  4
#define CHUNKS  (NSEQ / SPLITK / 64)   // 8 chunks of 64 rows per block
#define LN_EPSF 1e-5f

__device__ __forceinline__ v8f wmma_f32(v2f a, v2f b, v8f c) {
  // D(16x16) += A(16x4) * B(4x16), all f32.
  return __builtin_amdgcn_wmma_f32_16x16x4_f32(false, a, false, b, (short)0, c,
                                               false, false);
}

// ---------------- Tensor Data Mover availability / arity -------------------
#if defined(__has_builtin)
#  if __has_builtin(__builtin_amdgcn_tensor_load_to_lds) && \
      __has_builtin(__builtin_amdgcn_s_wait_tensorcnt)
#    define HAVE_TDM 1
#  endif
#endif
#ifndef HAVE_TDM
#  define HAVE_TDM 0
#endif
#if HAVE_TDM
#  if __has_include(<hip/amd_detail/amd_gfx1250_TDM.h>)
#    define TDM_ARGS 6   // therock-10.0 headers -> clang-23 6-arg builtin
#  else
#    define TDM_ARGS 5   // ROCm 7.2 clang-22 5-arg builtin
#  endif

typedef unsigned int u32x4v __attribute__((ext_vector_type(4)));
typedef int          i32x4v __attribute__((ext_vector_type(4)));
typedef int          i32x8v __attribute__((ext_vector_type(8)));

// Issue a TDM load of a 64x64 f32 tile (row stride 64) global -> LDS.
// D# layout per CDNA5 ISA ch.8: group0 {count|lds_addr|global_addr|type=2},
// group1 {data_size=4B, tensor_dim0/1=64, tile_dim0/1=64, dim0_stride=64}.
__device__ __forceinline__ void tdm_load_tile(const float* gsrc,
                                              unsigned lds_byte_off) {
  unsigned long long ga = (unsigned long long)(uintptr_t)gsrc;
  u32x4v g0;
  g0.x = 1u;                                     // count=1 (valid user D#)
  g0.y = lds_byte_off;                           // lds_addr
  g0.z = (unsigned)ga;                           // global_addr[31:0]
  g0.w = (unsigned)((ga >> 32) & 0x1FFFFFFull)   // global_addr[56:32]
         | (2u << 30);                           // type=2 ("image")
  i32x8v g1;
  g1[0] = 0x00020000;                            // data_size=2 -> 4 bytes
  g1[1] = (int)(64u << 16);                      // tensor_dim0 = 64
  g1[2] = (int)(64u << 16);                      // tensor_dim1 = 64
  g1[3] = (int)(64u << 16);                      // tile_dim0   = 64
  g1[4] = 64;                                    // tile_dim1   = 64
  g1[5] = 64;                                    // tensor_dim0_stride = 64
  g1[6] = 0;
  g1[7] = 0;
  i32x4v z4 = {0, 0, 0, 0};
#  if TDM_ARGS == 6
  i32x8v z8 = {0, 0, 0, 0, 0, 0, 0, 0};
  __builtin_amdgcn_tensor_load_to_lds(g0, g1, z4, z4, z8, 0);
#  else
  __builtin_amdgcn_tensor_load_to_lds(g0, g1, z4, z4, 0);
#  endif
}
#endif  // HAVE_TDM

// ===========================================================================
// Kernel 1: partial Gram G_s[bh] = H_s^T H_s over a 512-row K slice.
// grid (BH, SPLITK); 128 threads = 4 waves; wave w owns row-tile I=w, J=0..3.
// Double-buffered LDS staging: TDM DMA overlapped with WMMA compute.
// ===========================================================================
__global__ __launch_bounds__(128) void gram_kernel(const float* __restrict__ hist,
                                                   float* __restrict__ Gpart) {
  __shared__ float Hs[2][64][64];
  const int bh = blockIdx.x;
  const int split = blockIdx.y;
  const float* h = hist + ((size_t)bh * NSEQ + (size_t)split * (NSEQ / SPLITK)) * DH;
  float* g = Gpart + ((size_t)split * BH + bh) * DH * DH;

  const int tid = threadIdx.x;
  const int wave = tid >> 5, lane = tid & 31;
  const int I = wave;                 // output row tile
  const int ncol = lane & 15;         // M for A-frag, N for B-frag
  const int kb = (lane >> 4) << 1;    // K sub-offset per half-wave

  v8f acc[4] = {};
  int buf = 0;

#if HAVE_TDM
  const unsigned lds0 = (unsigned)(uintptr_t)&Hs[0][0][0];
  const unsigned lds1 = (unsigned)(uintptr_t)&Hs[1][0][0];
  if (tid < 32) tdm_load_tile(h, lds0);  // preload chunk 0 (wave 0 issues DMA)

  for (int chunk = 0; chunk < CHUNKS; ++chunk) {
    if (tid < 32) {
      if (chunk + 1 < CHUNKS) {
        tdm_load_tile(h + (size_t)(chunk + 1) * 64 * DH, buf ? lds0 : lds1);
        __builtin_amdgcn_s_wait_tensorcnt((short)1);  // chunk done, next inflight
      } else {
        __builtin_amdgcn_s_wait_tensorcnt((short)0);
      }
    }
    __syncthreads();  // LDS tile visible to all waves

    const float(*H)[64] = Hs[buf];
#pragma unroll 4
    for (int kk = 0; kk < 64; kk += 4) {
      v2f a;  // A[m][k] = hist[k][16I+m]
      a.x = H[kk + kb + 0][I * 16 + ncol];
      a.y = H[kk + kb + 1][I * 16 + ncol];
#pragma unroll
      for (int J = 0; J < 4; ++J) {
        v2f b;  // B[k][n] = hist[k][16J+n]
        b.x = H[kk + kb + 0][J * 16 + ncol];
        b.y = H[kk + kb + 1][J * 16 + ncol];
        acc[J] = wmma_f32(a, b, acc[J]);
      }
    }
    __syncthreads();  // all reads done before TDM may overwrite this buffer
    buf ^= 1;
  }
#else
  // Fallback: manual double-buffered staging + global_prefetch.
  {
    const float4* src = (const float4*)h;
#pragma unroll
    for (int it = 0; it < 8; ++it) {
      int idx = tid + it * 128;
      ((float4*)&Hs[0][idx >> 4][0])[idx & 15] = src[idx];
    }
  }
  __syncthreads();
  for (int chunk = 0; chunk < CHUNKS; ++chunk) {
    if (chunk + 1 < CHUNKS) {
      const float4* src = (const float4*)(h + (size_t)(chunk + 1) * 64 * DH);
#pragma unroll
      for (int it = 0; it < 8; ++it) {
        int idx = tid + it * 128;
        ((float4*)&Hs[buf ^ 1][idx >> 4][0])[idx & 15] = src[idx];
      }
      if (chunk + 2 < CHUNKS)
        __builtin_prefetch(h + (size_t)(chunk + 2) * 64 * DH + tid * 32, 0, 3);
    }
    const float(*H)[64] = Hs[buf];
#pragma unroll 4
    for (int kk = 0; kk < 64; kk += 4) {
      v2f a;
      a.x = H[kk + kb + 0][I * 16 + ncol];
      a.y = H[kk + kb + 1][I * 16 + ncol];
#pragma unroll
      for (int J = 0; J < 4; ++J) {
        v2f b;
        b.x = H[kk + kb + 0][J * 16 + ncol];
        b.y = H[kk + kb + 1][J * 16 + ncol];
        acc[J] = wmma_f32(a, b, acc[J]);
      }
    }
    __syncthreads();
    buf ^= 1;
  }
#endif

#pragma unroll
  for (int J = 0; J < 4; ++J)
#pragma unroll
    for (int r = 0; r < 8; ++r) {
      int row = I * 16 + ((lane >> 4) << 3) + r;  // C/D layout
      g[row * DH + J * 16 + ncol] = acc[J][r];
    }
}

// ===========================================================================
// Kernel 2: sum split-K partials, then symmetric Jacobi eigensolver on 64x64;
// emit bulk = sqrt(lambda)*v^T for the top-32 eigenpairs. 64 thr (2 waves).
// Parallel ordering: 63 rounds of 32 disjoint Givens rotations (circle method)
// ===========================================================================
__global__ __launch_bounds__(64) void jacobi_kernel(const float* __restrict__ Gpart,
                                                    float* __restrict__ bulk) {
  __shared__ float A[64][65];   // padded: column walks are conflict-free
  __shared__ float V[64][65];
  __shared__ float cs[32], sn[32];
  __shared__ int pp[32], qq[32];
  __shared__ float lam[64];

  const int bh = blockIdx.x;
  const int t = threadIdx.x;
  const float* g0 = Gpart + (size_t)bh * DH * DH;

  for (int i = t; i < 64 * 64; i += 64) {
    float s = 0.f;
#pragma unroll
    for (int sp = 0; sp < SPLITK; ++sp)
      s += g0[(size_t)sp * BH * DH * DH + i];
    A[i >> 6][i & 63] = s;
    V[i >> 6][i & 63] = ((i >> 6) == (i & 63)) ? 1.f : 0.f;
  }
  __syncthreads();

  for (int sweep = 0; sweep < 12; ++sweep) {
    for (int round = 0; round < 63; ++round) {
      if (t < 32) {
        int p, q;
        if (t == 0) { p = 63; q = round; }
        else { p = (round + t) % 63; q = (round - t + 63) % 63; }
        if (p > q) { int tmp = p; p = q; q = tmp; }
        float apq = A[p][q], app = A[p][p], aqq = A[q][q];
        float c = 1.f, s = 0.f;
        if (fabsf(apq) > 1e-20f) {
          float tau = (aqq - app) / (2.f * apq);
          float tt = copysignf(1.f, tau) / (fabsf(tau) + sqrtf(1.f + tau * tau));
          c = rsqrtf(1.f + tt * tt);
          s = tt * c;
        }
        cs[t] = c; sn[t] = s; pp[t] = p; qq[t] = q;
      }
      __syncthreads();
      // column update A <- A*J and V <- V*J ; thread owns row t
#pragma unroll 4
      for (int i = 0; i < 32; ++i) {
        int p = pp[i], q = qq[i];
        float c = cs[i], s = sn[i];
        float gp = A[t][p], gq = A[t][q];
        A[t][p] = c * gp - s * gq;
        A[t][q] = s * gp + c * gq;
        float vp = V[t][p], vq = V[t][q];
        V[t][p] = c * vp - s * vq;
        V[t][q] = s * vp + c * vq;
      }
      __syncthreads();
      // row update A <- J^T*A ; thread owns column t
#pragma unroll 4
      for (int i = 0; i < 32; ++i) {
        int p = pp[i], q = qq[i];
        float c = cs[i], s = sn[i];
        float rp = A[p][t], rq = A[q][t];
        A[p][t] = c * rp - s * rq;
        A[q][t] = s * rp + c * rq;
      }
      __syncthreads();
    }
  }

  lam[t] = A[t][t];
  __syncthreads();
  // rank of eigenvalue t (descending, index tie-break) -> select top 32
  float lj = lam[t];
  int rank = 0;
  for (int i = 0; i < 64; ++i) {
    float li = lam[i];
    rank += (li > lj) || (li == lj && i < t);
  }
  if (rank < KMODES) {
    float sv = sqrtf(fmaxf(lj, 0.f));
    float* brow = bulk + ((size_t)bh * KMODES + rank) * DH;
    for (int d = 0; d < DH; ++d) brow[d] = sv * V[d][t];  // Vh row = column t
  }
}

// ===========================================================================
// Kernel 3: kk = bulk @ Wk^T, vv = bulk @ Wv^T   (32x64 each, per head)
// waves 0,1 -> kk (row tiles 0,1); waves 2,3 -> vv.
// ===========================================================================
__global__ __launch_bounds__(128) void kv_kernel(const float* __restrict__ bulk,
                                                 const float* __restrict__ Wk,
                                                 const float* __restrict__ Wv,
                                                 float* __restrict__ kk,
                                                 float* __restrict__ vv) {
  __shared__ float Bs[KMODES][DH + 1];
  const int bh = blockIdx.x;
  const int tid = threadIdx.x, wave = tid >> 5, lane = tid & 31;
  const int ncol = lane & 15, kb = (lane >> 4) << 1;

  const float* bsrc = bulk + (size_t)bh * KMODES * DH;
  for (int i = tid; i < KMODES * DH; i += 128) Bs[i >> 6][i & 63] = bsrc[i];
  __syncthreads();

  const float* W = (wave < 2) ? Wk : Wv;
  float* dst = ((wave < 2) ? kk : vv) + (size_t)bh * KMODES * DH;
  const int I = wave & 1;

  v8f acc[4] = {};
  for (int k0 = 0; k0 < DH; k0 += 4) {
    v2f a;  // A[j][d] = bulk[16I+j][d]
    a.x = Bs[I * 16 + ncol][k0 + kb];
    a.y = Bs[I * 16 + ncol][k0 + kb + 1];
#pragma unroll
    for (int J = 0; J < 4; ++J) {
      v2f b;  // B[d][e] = W[e][d]
      const float* wr = W + (J * 16 + ncol) * DH + k0 + kb;
      b.x = wr[0];
      b.y = wr[1];
      acc[J] = wmma_f32(a, b, acc[J]);
    }
  }
#pragma unroll
  for (int J = 0; J < 4; ++J)
#pragma unroll
    for (int r = 0; r < 8; ++r)
      dst[(I * 16 + ((lane >> 4) << 3) + r) * DH + J * 16 + ncol] = acc[J][r];
}

// ===========================================================================
// Kernel 4: fused attention read + output proj + residual + LayerNorm.
// grid (BH, 8); 4 waves/block; wave owns one 16-row tile of current_state.
// ===========================================================================
__global__ __launch_bounds__(128)
void attn_kernel(const float* __restrict__ x, const float* __restrict__ kkg,
                 const float* __restrict__ vvg, const float* __restrict__ Wq,
                 const float* __restrict__ Wo, const float* __restrict__ bo,
                 const float* __restrict__ gam, const float* __restrict__ bet,
                 float* __restrict__ y) {
  __shared__ float KK[KMODES][DH + 1];
  __shared__ float VV[KMODES][DH + 1];
  __shared__ float Qs[4][16][DH + 1];      // q, later ctx
  __shared__ float Ss[4][16][KMODES + 1];  // scores -> attn
  __shared__ float Os[4][16][DH + 1];      // out projection

  const int bh = blockIdx.x;
  const int tid = threadIdx.x, wave = tid >> 5, lane = tid & 31;
  const int ncol = lane & 15, kb = (lane >> 4) << 1;
  const int rrow = ((lane >> 4) << 3);     // C/D row offset per half-wave
  const int m0 = blockIdx.y * 64 + wave * 16;
  const float* xrow = x + ((size_t)bh * MROWS + m0) * DH;

  {
    const float* ks = kkg + (size_t)bh * KMODES * DH;
    const float* vs = vvg + (size_t)bh * KMODES * DH;
    for (int i = tid; i < KMODES * DH; i += 128) {
      KK[i >> 6][i & 63] = ks[i];
      VV[i >> 6][i & 63] = vs[i];
    }
  }
  __syncthreads();

  // ---- q = x @ Wq^T ----
  v8f qacc[4] = {};
  for (int d0 = 0; d0 < DH; d0 += 4) {
    v2f a;
    a.x = xrow[ncol * DH + d0 + kb];
    a.y = xrow[ncol * DH + d0 + kb + 1];
#pragma unroll
    for (int E = 0; E < 4; ++E) {
      v2f b;
      b.x = Wq[(E * 16 + ncol) * DH + d0 + kb];
      b.y = Wq[(E * 16 + ncol) * DH + d0 + kb + 1];
      qacc[E] = wmma_f32(a, b, qacc[E]);
    }
  }
#pragma unroll
  for (int E = 0; E < 4; ++E)
#pragma unroll
    for (int r = 0; r < 8; ++r) Qs[wave][rrow + r][E * 16 + ncol] = qacc[E][r];
  __syncthreads();

  // ---- scores = q @ kk^T * (1/sqrt(Dh)) ----
  v8f sacc[2] = {};
  for (int e0 = 0; e0 < DH; e0 += 4) {
    v2f a;
    a.x = Qs[wave][ncol][e0 + kb];
    a.y = Qs[wave][ncol][e0 + kb + 1];
#pragma unroll
    for (int J = 0; J < 2; ++J) {
      v2f b;  // B[e][j] = kk[16J+j][e]
      b.x = KK[J * 16 + ncol][e0 + kb];
      b.y = KK[J * 16 + ncol][e0 + kb + 1];
      sacc[J] = wmma_f32(a, b, sacc[J]);
    }
  }
#pragma unroll
  for (int J = 0; J < 2; ++J)
#pragma unroll
    for (int r = 0; r < 8; ++r)
      Ss[wave][rrow + r][J * 16 + ncol] = sacc[J][r] * 0.125f;
  __syncthreads();

  // ---- softmax over 32 modes (one lane per row) ----
  if (lane < 16) {
    float mx = -3.4e38f;
    for (int j = 0; j < KMODES; ++j) mx = fmaxf(mx, Ss[wave][lane][j]);
    float sum = 0.f;
    for (int j = 0; j < KMODES; ++j) {
      float e = __expf(Ss[wave][lane][j] - mx);
      Ss[wave][lane][j] = e;
      sum += e;
    }
    float inv = 1.f / sum;
    for (int j = 0; j < KMODES; ++j) Ss[wave][lane][j] *= inv;
  }
  __syncthreads();

  // ---- ctx = attn @ vv ----
  v8f cacc[4] = {};
  for (int j0 = 0; j0 < KMODES; j0 += 4) {
    v2f a;
    a.x = Ss[wave][ncol][j0 + kb];
    a.y = Ss[wave][ncol][j0 + kb + 1];
#pragma unroll
    for (int E = 0; E < 4; ++E) {
      v2f b;
      b.x = VV[j0 + kb + 0][E * 16 + ncol];
      b.y = VV[j0 + kb + 1][E * 16 + ncol];
      cacc[E] = wmma_f32(a, b, cacc[E]);
    }
  }
#pragma unroll
  for (int E = 0; E < 4; ++E)
#pragma unroll
    for (int r = 0; r < 8; ++r) Qs[wave][rrow + r][E * 16 + ncol] = cacc[E][r];
  __syncthreads();

  // ---- out = ctx @ Wo^T ----
  v8f oacc[4] = {};
  for (int d0 = 0; d0 < DH; d0 += 4) {
    v2f a;
    a.x = Qs[wave][ncol][d0 + kb];
    a.y = Qs[wave][ncol][d0 + kb + 1];
#pragma unroll
    for (int E = 0; E < 4; ++E) {
      v2f b;
      b.x = Wo[(E * 16 + ncol) * DH + d0 + kb];
      b.y = Wo[(E * 16 + ncol) * DH + d0 + kb + 1];
      oacc[E] = wmma_f32(a, b, oacc[E]);
    }
  }
#pragma unroll
  for (int E = 0; E < 4; ++E)
#pragma unroll
    for (int r = 0; r < 8; ++r) Os[wave][rrow + r][E * 16 + ncol] = oacc[E][r];
  __syncthreads();

  // ---- residual + LayerNorm (one lane per row) ----
  if (lane < 16) {
    const float* xr = xrow + lane * DH;
    float sum = 0.f, sumsq = 0.f;
    for (int e = 0; e < DH; ++e) {
      float v = xr[e] + Os[wave][lane][e] + bo[e];
      sum += v;
      sumsq += v * v;
    }
    float mu = sum * (1.f / DH);
    float var = sumsq * (1.f / DH) - mu * mu;
    float rstd = rsqrtf(var + LN_EPSF);
    float* yr = y + ((size_t)bh * MROWS + m0 + lane) * DH;
    for (int e = 0; e < DH; ++e) {
      float v = xr[e] + Os[wave][lane][e] + bo[e];
      yr[e] = (v - mu) * rstd * gam[e] + bet[e];
    }
  }
}

// ===========================================================================
extern "C" void kernel_launch(void* const* d_in, const int* in_sizes, int n_in,
                              void* d_out, int out_size, void* d_ws,
                              size_t ws_size, hipStream_t stream) {
  (void)in_sizes; (void)n_in; (void)out_size; (void)ws_size;
  const float* x    = (const float*)d_in[0];  // current_state (8,16,512,64)
  const float* hist = (const float*)d_in[1];  // history       (8,16,2048,64)
  const float* Wq   = (const float*)d_in[2];
  const float* Wk   = (const float*)d_in[3];
  const float* Wv   = (const float*)d_in[4];
  const float* Wo   = (const float*)d_in[5];
  const float* bo   = (const float*)d_in[6];
  const float* gam  = (const float*)d_in[7];
  const float* bet  = (const float*)d_in[8];
  float* out = (float*)d_out;

  // workspace layout (~11.5 MB total)
  float* Gpart = (float*)d_ws;                          // SPLITK*128*64*64
  float* bulk  = Gpart + (size_t)SPLITK * BH * DH * DH; // 128*32*64
  float* kk    = bulk + (size_t)BH * KMODES * DH;
  float* vv    = kk + (size_t)BH * KMODES * DH;

  gram_kernel<<<dim3(BH, SPLITK), dim3(128), 0, stream>>>(hist, Gpart);
  jacobi_kernel<<<dim3(BH), dim3(64), 0, stream>>>(Gpart, bulk);
  kv_kernel<<<dim3(BH), dim3(128), 0, stream>>>(bulk, Wk, Wv, kk, vv);
  attn_kernel<<<dim3(BH, 8), dim3(128), 0, stream>>>(x, kk, vv, Wq, Wo, bo,
                                                     gam, bet, out);
}